// positional_encoding_grid_64149631533762
// MI455X (gfx1250) — compile-verified
//
#include <hip/hip_runtime.h>
#include <stdint.h>

// ---------------- HashGrid config (matches reference) ----------------
#define NLVL      16
#define TBL_T     (1u << 19)           // 524288 entries per level
#define TBL_MASK  (TBL_T - 1u)
#define P1        1u
#define P2        2654435761u
#define P3        805459861u

// Per-level scale: 16 * s^l - 1  (double at compile time, cast to f32 like the reference)
constexpr double PLS = 1.4472692012786865;
constexpr double pls_pow(int l) { double r = 1.0; for (int i = 0; i < l; ++i) r *= PLS; return r; }
constexpr float  lvl_scale(int l) { return (float)(16.0 * pls_pow(l) - 1.0); }

__device__ __constant__ float kScale[NLVL] = {
  lvl_scale(0),  lvl_scale(1),  lvl_scale(2),  lvl_scale(3),
  lvl_scale(4),  lvl_scale(5),  lvl_scale(6),  lvl_scale(7),
  lvl_scale(8),  lvl_scale(9),  lvl_scale(10), lvl_scale(11),
  lvl_scale(12), lvl_scale(13), lvl_scale(14), lvl_scale(15)
};
// res = ceil(scale)+1 ; only levels 0..4 dense (res^3 <= T), rest hashed (res unused there)
__device__ __constant__ uint32_t kRes[NLVL] = {
  16u, 24u, 34u, 49u, 71u, 102u, 148u, 213u, 308u, 446u, 646u, 934u, 1352u, 1956u, 2831u, 4097u
};

#define L0_ELEMS 4096                      // 16^3 entries -> 32 KB LDS (keeps occupancy high)

// Native clang vector types (HIP float4 is a class; builtins need real vectors)
typedef float f32x4 __attribute__((ext_vector_type(4)));

// ---------------- CDNA5 async global->LDS helpers ----------------
typedef int v2i __attribute__((vector_size(2 * sizeof(int))));
typedef __attribute__((address_space(1))) v2i* as1_v2i;
typedef __attribute__((address_space(3))) v2i* as3_v2i;

__device__ __forceinline__ void async_copy_f2(float2* dst_lds, const float2* src_g) {
#if __has_builtin(__builtin_amdgcn_global_load_async_to_lds_b64)
  __builtin_amdgcn_global_load_async_to_lds_b64(
      (as1_v2i)(void*)const_cast<float2*>(src_g),
      (as3_v2i)(void*)dst_lds,
      /*offset=*/0, /*cpol=*/0);
#else
  unsigned lds_off = (unsigned)(unsigned long long)dst_lds;
  asm volatile("global_load_async_to_lds_b64 %0, %1, off"
               :: "v"(lds_off), "v"(src_g) : "memory");
#endif
}

__device__ __forceinline__ void wait_async0() {
#if __has_builtin(__builtin_amdgcn_s_wait_asynccnt)
  __builtin_amdgcn_s_wait_asynccnt(0);
#else
  asm volatile("s_wait_asynccnt 0" ::: "memory");
#endif
}

// ---------------- per-level encode ----------------
__device__ __forceinline__ float2 encode_level(const float2* __restrict__ gtab,
                                               const float2* __restrict__ ltab,
                                               bool dense, bool use_lds,
                                               float scale, uint32_t res,
                                               float x, float y, float z) {
  float px = fmaf(x, scale, 0.5f);
  float py = fmaf(y, scale, 0.5f);
  float pz = fmaf(z, scale, 0.5f);
  float fx = floorf(px), fy = floorf(py), fz = floorf(pz);
  float wx = px - fx,   wy = py - fy,   wz = pz - fz;
  uint32_t ix = (uint32_t)fx, iy = (uint32_t)fy, iz = (uint32_t)fz;

  // Per-dimension index contributions hoisted out of the 8-corner loop:
  // dense: clamped coords scaled by {1, res, res^2}; hashed: prime products.
  uint32_t hx[2], hy[2], hz[2];
  if (dense) {
    uint32_t m = res - 1u;
    uint32_t cx0 = ix < m ? ix : m, cx1 = (ix + 1u) < m ? (ix + 1u) : m;
    uint32_t cy0 = iy < m ? iy : m, cy1 = (iy + 1u) < m ? (iy + 1u) : m;
    uint32_t cz0 = iz < m ? iz : m, cz1 = (iz + 1u) < m ? (iz + 1u) : m;
    hx[0] = cx0;             hx[1] = cx1;
    hy[0] = cy0 * res;       hy[1] = cy1 * res;
    hz[0] = cz0 * res * res; hz[1] = cz1 * res * res;
  } else {
    hx[0] = ix * P1; hx[1] = (ix + 1u) * P1;
    hy[0] = iy * P2; hy[1] = (iy + 1u) * P2;
    hz[0] = iz * P3; hz[1] = (iz + 1u) * P3;
  }

  float2 v[8];
#pragma unroll
  for (int k = 0; k < 8; ++k) {       // OFFSETS[k] = (k>>2, k>>1&1, k&1), meshgrid 'ij'
    int dx = (k >> 2) & 1, dy = (k >> 1) & 1, dz = k & 1;
    uint32_t idx = dense ? (hx[dx] + hy[dy] + hz[dz])
                         : ((hx[dx] ^ hy[dy] ^ hz[dz]) & TBL_MASK);
    v[k] = use_lds ? ltab[idx] : gtab[idx];   // ds_load_b64 or global_load_b64
  }

  float wx0 = 1.0f - wx, wy0 = 1.0f - wy, wz0 = 1.0f - wz;
  float2 acc; acc.x = 0.0f; acc.y = 0.0f;
#pragma unroll
  for (int k = 0; k < 8; ++k) {
    float w = (((k >> 2) & 1) ? wx : wx0) *
              (((k >> 1) & 1) ? wy : wy0) *
              (((k     ) & 1) ? wz : wz0);
    acc.x = fmaf(w, v[k].x, acc.x);
    acc.y = fmaf(w, v[k].y, acc.y);
  }
  return acc;
}

// ---------------- main kernel ----------------
__launch_bounds__(256)
__global__ void hashgrid_encode_kernel(const float* __restrict__ coords,
                                       const float* __restrict__ table,
                                       float* __restrict__ out,
                                       int nPoints) {
  __shared__ float2 sh[L0_ELEMS];              // level-0 dense table, 32 KB
  const float2* gt = (const float2*)table;     // level l base = gt + l*TBL_T

  // Stage dense level 0 into LDS with CDNA5 async copies (amortized over ~512 pts/block).
  for (int e = (int)threadIdx.x; e < L0_ELEMS; e += (int)blockDim.x)
    async_copy_f2(&sh[e], gt + e);
  wait_async0();
  __syncthreads();

  const int stride = (int)(gridDim.x * blockDim.x);
  for (int n = (int)(blockIdx.x * blockDim.x + threadIdx.x); n < nPoints; n += stride) {
    // Read-once coords: non-temporal so the 12 MB stream doesn't pollute L2.
    float x = __builtin_nontemporal_load(coords + 3 * n + 0);
    float y = __builtin_nontemporal_load(coords + 3 * n + 1);
    float z = __builtin_nontemporal_load(coords + 3 * n + 2);

    float acc[2 * NLVL];
#pragma unroll
    for (int l = 0; l < NLVL; ++l) {
      const bool dense   = (l < 5);
      const bool use_lds = (l < 1);
      const float2* gtab = gt + (size_t)l * TBL_T;
      float2 f = encode_level(gtab, sh, dense, use_lds, kScale[l], kRes[l], x, y, z);
      acc[2 * l + 0] = f.x;
      acc[2 * l + 1] = f.y;
    }

    // Write-once 128 MB output: non-temporal b128 stores keep the 64 MB of
    // hash tables resident in the 192 MB L2 (gathers are the bottleneck).
    f32x4* o = (f32x4*)(out + (size_t)n * (2 * NLVL));
#pragma unroll
    for (int j = 0; j < 8; ++j) {
      f32x4 q = { acc[4 * j + 0], acc[4 * j + 1], acc[4 * j + 2], acc[4 * j + 3] };
      __builtin_nontemporal_store(q, o + j);
    }
  }
}

extern "C" void kernel_launch(void* const* d_in, const int* in_sizes, int n_in,
                              void* d_out, int out_size, void* d_ws, size_t ws_size,
                              hipStream_t stream) {
  (void)n_in; (void)out_size; (void)d_ws; (void)ws_size;
  const float* coords = (const float*)d_in[0];
  const float* table  = (const float*)d_in[1];
  float* out          = (float*)d_out;
  int nPoints = in_sizes[0] / 3;

  const int block  = 256;   // 8 waves (wave32)
  const int blocks = 2048;  // 2 points/thread grid-stride; 32 KB stage amortized over 512 pts
  hashgrid_encode_kernel<<<blocks, block, 0, stream>>>(coords, table, out, nPoints);
}